// AtomPoolingLayer_34926674051615
// MI455X (gfx1250) — compile-verified
//
#include <hip/hip_runtime.h>

// AtomPoolingLayer for MI455X (gfx1250, wave32, WMMA + async-to-LDS).
// h:[512,128,512] f32, W1:[512,128], b1:[128], W2:[128,1], b2:[1]
// out:[512,512] f32
//
// One block per molecule, 256 threads = 8 wave32.
// Phase 1: T = relu(h_m @ W1 + b1) via v_wmma_f32_16x16x4_f32.
//          W1 staged into LDS in double-buffered 128-row K-panels using
//          global_load_async_to_lds_b32 (ASYNCcnt), stored K-pair-interleaved
//          so each B fragment is one aligned ds_load_b64 (no repack movs).
// Phase 2: gate = sigmoid(T @ W2 + b2) through LDS.
// Phase 3: out_m = gate^T @ h_m (h re-read; L2-resident: 134MB h < 192MB L2).

typedef __attribute__((ext_vector_type(2))) float v2f;
typedef __attribute__((ext_vector_type(8))) float v8f;

#define MOLS     512
#define NATOMS   128
#define FDIM     512
#define HIDDEN   128
#define TSTRIDE  132            // tT row pad: avoid bank aliasing
#define PANEL_K  128            // K-rows of W1 per LDS panel
#define NPANEL   (FDIM / PANEL_K)
// Interleaved layout: element (k,c) at float offset (k>>1)*PAIRROWF + 2*c + (k&1).
// PAIRROWF = 2*HIDDEN + 32 pad: 288 mod 64 == 32 -> the two lane-halves of a
// fragment read disjoint 32-bank halves; within a half, stride-2 covers 32 banks.
#define PAIRROWF     288
#define PANEL_FLOATS ((PANEL_K / 2) * PAIRROWF)   // 64 pair-rows * 288 = 18432
#define PANEL_BYTES  (PANEL_FLOATS * 4)           // 73728 B

// Stage one 128x128 K-panel of W1 into LDS, K-pair-interleaved, via async b32
// scatter. Global side stays fully coalesced (consecutive tid -> consecutive
// floats); LDS side is stride-2 (conflict-free banks). 64 ops/thread.
__device__ __forceinline__ void stage_panel_async(const float* __restrict__ W1,
                                                  unsigned lds_base_bytes,
                                                  int p, int tid) {
    #pragma unroll 8
    for (int i = 0; i < 64; ++i) {
        const int e  = i * 256 + tid;           // flat element in 128x128 panel
        const int kl = e >> 7;                  // k row within panel
        const int c  = e & 127;                 // hidden column
        unsigned lds_addr = lds_base_bytes +
            (unsigned)((kl >> 1) * PAIRROWF + 2 * c + (kl & 1)) * 4u;
        unsigned voff = (unsigned)(p * (PANEL_K * HIDDEN) + e) * 4u;
        asm volatile("global_load_async_to_lds_b32 %0, %1, %2"
                     :: "v"(lds_addr), "v"(voff), "s"(W1)
                     : "memory");
    }
}

__global__ __launch_bounds__(256)
void atom_pool_wmma_kernel(const float* __restrict__ h,
                           const float* __restrict__ W1,
                           const float* __restrict__ b1,
                           const float* __restrict__ W2,
                           const float* __restrict__ b2,
                           float* __restrict__ out)
{
    extern __shared__ float lds[];
    // Phase 1: lds = two interleaved W1 panels (double buffer).
    // Phase 2/3 (aliased after barrier): tT[128][TSTRIDE] + wgate[128].
    float* tT    = lds;
    float* wgate = lds + NATOMS * TSTRIDE;

    const int m    = blockIdx.x;
    const int tid  = threadIdx.x;
    const int wave = tid >> 5;   // 0..7 : row strip 16*wave .. 16*wave+15
    const int lane = tid & 31;
    const int r    = lane & 15;  // row-in-strip (A) / col-in-tile (B,C)
    const int half = lane >> 4;  // K-half select for 16x16x4 fragments

    const unsigned lds_base = (unsigned)(uintptr_t)lds;  // LDS byte offset (addr[31:0])
    const float* __restrict__ hm = h + (size_t)m * NATOMS * FDIM;

    // ---------------- Phase 1: T strip = relu(h_m @ W1 + b1) ----------------
    v8f acc[8] = {};

    // A fragment (16x4 f32): lane l<16 -> M=l, K={k,k+1}; l>=16 -> M=l-16, K={k+2,k+3}
    const float* __restrict__ arow = hm + (size_t)(16 * wave + r) * FDIM + 2 * half;

    // Prime panel 0.
    stage_panel_async(W1, lds_base, 0, tid);
    asm volatile("s_wait_asynccnt 0x0" ::: "memory");
    __syncthreads();

    for (int p = 0; p < NPANEL; ++p) {
        if (p + 1 < NPANEL)
            stage_panel_async(W1, lds_base + ((p + 1) & 1) * PANEL_BYTES, p + 1, tid);

        const float* __restrict__ buf = lds + (p & 1) * PANEL_FLOATS;
        const int kbase = p * PANEL_K;

        for (int kk = 0; kk < PANEL_K; kk += 4) {
            v2f a = *(const v2f*)(arow + kbase + kk);          // 8B-aligned
            __builtin_prefetch(arow + kbase + kk + 128, 0, 3); // stream h ahead

            // B fragment (4x16 f32) for tile t: one aligned b64 slot holding
            // {W1[kk+2h][c], W1[kk+2h+1][c]} at wb + 32*t (imm offset 128*t B).
            const float* __restrict__ wb =
                buf + ((kk >> 1) + half) * PAIRROWF + 2 * r;
            #pragma unroll
            for (int t = 0; t < 8; ++t) {
                v2f b = *(const v2f*)(wb + 32 * t);
                acc[t] = __builtin_amdgcn_wmma_f32_16x16x4_f32(
                    /*neg_a=*/false, a, /*neg_b=*/false, b,
                    /*c_mod=*/(short)0, acc[t],
                    /*reuse_a=*/false, /*reuse_b=*/false);
            }
        }
        // Next panel fully in LDS (own asyncs waited, others via barrier);
        // also guarantees all waves finished reading the buffer staged next.
        asm volatile("s_wait_asynccnt 0x0" ::: "memory");
        __syncthreads();
    }

    // Bias + ReLU, spill strip to LDS (aliases the dead W1 panels).
    // C/D 16x16 f32 layout: lane 0-15 -> N=lane, VGPR j -> M=j; lane 16-31 -> M=j+8.
    #pragma unroll
    for (int t = 0; t < 8; ++t) {
        const int col  = 16 * t + r;
        const float bv = b1[col];
        #pragma unroll
        for (int j = 0; j < 8; ++j) {
            const int rowl = j + (half ? 8 : 0);
            float v = acc[t][j] + bv;
            tT[(16 * wave + rowl) * TSTRIDE + col] = fmaxf(v, 0.0f);
        }
    }
    __syncthreads();

    // ---------------- Phase 2: gate = sigmoid(T @ W2 + b2) ----------------
    if (tid < NATOMS) {
        float s = b2[0];
        const float* trow = tT + tid * TSTRIDE;
        #pragma unroll 4
        for (int c = 0; c < HIDDEN; ++c) {
            s = fmaf(trow[c], W2[c], s);
        }
        wgate[tid] = 1.0f / (1.0f + __expf(-s));
    }
    __syncthreads();

    // ---------------- Phase 3: out_m[f] = sum_n gate[n] * h_m[n][f] ----------------
    // Coalesced float2: thread tid handles f = 2*tid, 2*tid+1. h is L2-resident here.
    float2 o = make_float2(0.0f, 0.0f);
    const float2* __restrict__ hp2 = (const float2*)hm + tid;
    #pragma unroll 4
    for (int n = 0; n < NATOMS; ++n) {
        const float wn = wgate[n];
        const float2 v = hp2[n * (FDIM / 2)];
        o.x = fmaf(wn, v.x, o.x);
        o.y = fmaf(wn, v.y, o.y);
    }
    ((float2*)out)[(size_t)m * (FDIM / 2) + tid] = o;
}

extern "C" void kernel_launch(void* const* d_in, const int* in_sizes, int n_in,
                              void* d_out, int out_size, void* d_ws, size_t ws_size,
                              hipStream_t stream) {
    const float* h  = (const float*)d_in[0];
    const float* W1 = (const float*)d_in[1];
    const float* b1 = (const float*)d_in[2];
    const float* W2 = (const float*)d_in[3];
    const float* b2 = (const float*)d_in[4];
    float* out = (float*)d_out;

    size_t lds_bytes = 2 * (size_t)PANEL_BYTES;  // panels; tT/gate alias inside
    const size_t phase2_bytes = (size_t)(NATOMS * TSTRIDE + NATOMS) * sizeof(float);
    if (phase2_bytes > lds_bytes) lds_bytes = phase2_bytes;

    atom_pool_wmma_kernel<<<MOLS, 256, lds_bytes, stream>>>(h, W1, b1, W2, b2, out);
}